// WaterNet1_76373108457548
// MI455X (gfx1250) — compile-verified
//
#include <hip/hip_runtime.h>
#include <math.h>

typedef __attribute__((ext_vector_type(2))) float v2f;
typedef __attribute__((ext_vector_type(8))) float v8f;

#define NT 1024
#define NS 2048
#define WAVES_PER_BLOCK 8
#define BLOCK_THREADS (WAVES_PER_BLOCK * 32)

// DPP8 selector: 3 bits per lane within each group of 8.
#define DPP8_SEL(a,b,c,d,e,f,g,h) \
  ((a)|((b)<<3)|((c)<<6)|((d)<<9)|((e)<<12)|((f)<<15)|((g)<<18)|((h)<<21))
constexpr int DPP8_XOR1 = DPP8_SEL(1,0,3,2,5,4,7,6);
constexpr int DPP8_XOR2 = DPP8_SEL(2,3,0,1,6,7,4,5);
constexpr int DPP8_XOR4 = DPP8_SEL(4,5,6,7,0,1,2,3);

// Broadcast a float from a (constant-after-unroll) lane -> v_readlane_b32 SGPR.
__device__ __forceinline__ float bcastf(float v, int l) {
  return __builtin_bit_cast(
      float, __builtin_amdgcn_readlane(__builtin_bit_cast(int, v), l));
}

// xor-4/2/1 lane swaps on the VALU via DPP8 (no LDS traffic).
template <int SEL>
__device__ __forceinline__ float dpp8f(float v) {
  return __builtin_bit_cast(
      float, __builtin_amdgcn_mov_dpp8(__builtin_bit_cast(int, v), SEL));
}

// xor-8 within each 16-lane row via v_permlane16_b32 (VALU, no LDS).
// Lane j in each row reads lane j^8: sel_lo = lanes0-7 <- 8..15, sel_hi = 0..7.
__device__ __forceinline__ float xor8f(float v) {
  int x = __builtin_bit_cast(int, v);
  int r = __builtin_amdgcn_permlane16(x, x, 0xFEDCBA98u, 0x76543210u,
                                      /*fi=*/false, /*bound_ctrl=*/false);
  return __builtin_bit_cast(float, r);
}

// First butterfly stage (xor-16) on the matrix pipe:
// A = all-ones 16x4, B = {p, 0} 4x16  =>  D[m,n] = p[n] + p[n+16].
// Any documented B/D layout maps lane L to column n = L%16, and all-ones A
// makes D[.,n] the column sum, so this is layout-robust. d[0] = p[L%16]+p[L%16+16].
__device__ __forceinline__ float wmma_fold16(float p) {
  v2f a; a.x = 1.0f; a.y = 1.0f;
  v2f b; b.x = p;    b.y = 0.0f;
  v8f c = {};
  v8f d = __builtin_amdgcn_wmma_f32_16x16x4_f32(
      /*neg_a=*/false, a, /*neg_b=*/false, b,
      /*c_mod=*/(short)0, c, /*reuse_a=*/false, /*reuse_b=*/false);
  return d[0];
}

__device__ __forceinline__ float sigmoidf(float x) {
  return 1.0f / (1.0f + expf(-x));
}

__global__ __launch_bounds__(BLOCK_THREADS) void waternet_scan(
    const float* __restrict__ P,  const float* __restrict__ T1,
    const float* __restrict__ T2, const float* __restrict__ E,
    const float* __restrict__ w,  float* __restrict__ out) {
  const int lane = threadIdx.x & 31;                   // h index
  const int wv   = threadIdx.x >> 5;                   // wave within block
  const int s    = blockIdx.x * WAVES_PER_BLOCK + wv;  // site

  // Per-hidden gates (lane = h); 128-float wave-broadcast loads.
  const float gm = expf(w[lane]) + 1.0f;         // melt factor
  const float ge = sigmoidf(w[32 + lane]);       // ET gate
  const float go = sigmoidf(w[64 + lane]);       // outflow gate
  // softmax over 32 lanes (max-subtracted, like jax.nn.softmax); one-time cost.
  const float xw = w[96 + lane];
  float mx = xw;
  for (int off = 16; off; off >>= 1) mx = fmaxf(mx, __shfl_xor(mx, off, 32));
  const float ex = expf(xw - mx);
  float sm = ex;
  for (int off = 16; off; off >>= 1) sm += __shfl_xor(sm, off, 32);
  const float ga = ex / sm;

  float S = 0.0f, H = 0.0f;

  for (int t0 = 0; t0 < NT; t0 += 32) {
    // Lane L precomputes the rain/snow partition for time step t0+L
    // (amortizes the acos transcendental 32x across the wave).
    const int idx = (t0 + lane) * NS + s;
    const float Pv  = P[idx];
    const float t1v = T1[idx];
    const float t2v = T2[idx];
    const float Ev  = E[idx];

    // Prefetch the next 32-step block of forcings.
    if (t0 + 32 < NT) {
      const int nidx = idx + 32 * NS;
      __builtin_prefetch(&P[nidx],  0, 0);
      __builtin_prefetch(&T1[nidx], 0, 0);
      __builtin_prefetch(&T2[nidx], 0, 0);
      __builtin_prefetch(&E[nidx],  0, 0);
    }

    const float Ta = (t1v + t2v) * 0.5f;
    float ratio = (t1v + t2v) / (t2v - t1v);
    ratio = fminf(1.0f, fmaxf(-1.0f, ratio));
    float rp = 1.0f - acosf(ratio) / 3.1415f;    // literal PI from reference
    rp = (t1v >= 0.0f) ? 1.0f : rp;
    rp = (t2v <= 0.0f) ? 0.0f : rp;
    const float Psn = (1.0f - rp) * Pv;          // snow
    const float Plq = rp * Pv;                   // liquid rain

    float yv = 0.0f;
#pragma unroll
    for (int k = 0; k < 32; ++k) {
      // Wave-uniform forcings for step t0+k (SGPR broadcasts).
      const float Ta_k = bcastf(Ta,  k);
      const float Ps_k = bcastf(Psn, k);
      const float Pl_k = bcastf(Plq, k);
      const float E_k  = bcastf(Ev,  k);

      // State update (per lane h).
      const float Smelt = fminf(S, fmaxf(Ta_k * gm, 0.0f));
      S = S + Ps_k - Smelt;                      // snow store
      float Hn = H + Smelt + Pl_k - E_k * ge;    // water store
      Hn = fmaxf(Hn, 0.0f);
      const float Q = Hn * go;                   // outflow
      H = Hn - Q;

      // Y = sum over 32 lanes of Q*ga:
      // xor16 on the matrix pipe, xor8 via permlane16, xor4/2/1 via DPP8.
      float r = wmma_fold16(Q * ga);
      r += xor8f(r);
      r += dpp8f<DPP8_XOR4>(r);
      r += dpp8f<DPP8_XOR2>(r);
      r += dpp8f<DPP8_XOR1>(r);

      yv = (lane == k) ? r : yv;  // lane k keeps Y for step t0+k
    }
    out[idx] = yv;                // lane L stores Y[t0+L, s]
  }
}

extern "C" void kernel_launch(void* const* d_in, const int* in_sizes, int n_in,
                              void* d_out, int out_size, void* d_ws, size_t ws_size,
                              hipStream_t stream) {
  (void)in_sizes; (void)n_in; (void)out_size; (void)d_ws; (void)ws_size;
  const float* P  = (const float*)d_in[0];
  const float* T1 = (const float*)d_in[1];
  const float* T2 = (const float*)d_in[2];
  const float* E  = (const float*)d_in[3];
  const float* w  = (const float*)d_in[4];
  float* out = (float*)d_out;

  dim3 grid(NS / WAVES_PER_BLOCK);   // 256 blocks -> 2048 wave32 waves
  dim3 block(BLOCK_THREADS);         // 256 threads = 8 waves
  waternet_scan<<<grid, block, 0, stream>>>(P, T1, T2, E, w, out);
}